// MessagePassing_18545668784703
// MI455X (gfx1250) — compile-verified
//
#include <hip/hip_runtime.h>
#include <hip/hip_bf16.h>

// Scatter-add message passing:
//   out[dst[e], :] += x[src[e], :]   for e in [0, E), D_FEAT = 32 (128B per row)
//
// Roofline: 51 MFLOP vs ~420MB of L2 traffic; x (6.4MB), out (6.4MB) and the
// index stream (12.8MB) all fit in the 192MB L2 -> HBM traffic ~25MB total.
// Limiter is L2 atomic throughput (51.2M f32 RMWs), so the kernel minimizes
// wave-level VMEM instructions per edge and uses fire-and-forget atomics:
//
//   8 lanes per edge, 2 edges per group iteration:
//     group: 1x b64 NT load (src pair) + 1x b64 NT load (dst pair)
//     per edge, per lane: 1x global_load_b128 (RT; x rows are reused ~32x,
//                          keep them in WGP$/L2) + 4x global_atomic_add_f32
//
// Index loads are non-temporal: read-once stream, don't evict reusable x rows
// from the near cache.

#define D_FEAT 32

typedef int v2i __attribute__((ext_vector_type(2)));

__global__ __launch_bounds__(256) void zero_out_f4(float4* __restrict__ out, int n4) {
    int i = blockIdx.x * blockDim.x + threadIdx.x;
    if (i < n4) {
        out[i] = make_float4(0.0f, 0.0f, 0.0f, 0.0f);
    }
}

__global__ __launch_bounds__(256) void scatter_add_edges(
    const float* __restrict__ x,
    const int* __restrict__ edge_index,  // [2, E] flat: row0 = src, row1 = dst
    float* __restrict__ out,
    int num_edges)
{
    const int gid  = blockIdx.x * blockDim.x + threadIdx.x;
    const int pair = gid >> 3;           // edge-pair id (8 lanes x 2 edges)
    const int r    = gid & 7;            // float4 chunk within the 32-float row
    const int e0   = pair << 1;          // first edge of the pair
    if (e0 >= num_edges) return;

    // Non-temporal b64 loads of the index pairs (streaming, read-once).
    // e0 and num_edges are both even -> 8B-aligned addresses.
    const v2i src2 = __builtin_nontemporal_load(
        reinterpret_cast<const v2i*>(edge_index + e0));
    const v2i dst2 = __builtin_nontemporal_load(
        reinterpret_cast<const v2i*>(edge_index + num_edges + e0));

    const int col = r << 2;              // float offset of this lane's chunk

    // ---- edge e0 ----
    {
        const float4 v = *reinterpret_cast<const float4*>(
            x + (size_t)src2.x * D_FEAT + col);
        float* o = out + (size_t)dst2.x * D_FEAT + col;
        unsafeAtomicAdd(o + 0, v.x);
        unsafeAtomicAdd(o + 1, v.y);
        unsafeAtomicAdd(o + 2, v.z);
        unsafeAtomicAdd(o + 3, v.w);
    }

    // ---- edge e0+1 (guard the tail when num_edges is odd) ----
    if (e0 + 1 < num_edges) {
        const float4 v = *reinterpret_cast<const float4*>(
            x + (size_t)src2.y * D_FEAT + col);
        float* o = out + (size_t)dst2.y * D_FEAT + col;
        unsafeAtomicAdd(o + 0, v.x);
        unsafeAtomicAdd(o + 1, v.y);
        unsafeAtomicAdd(o + 2, v.z);
        unsafeAtomicAdd(o + 3, v.w);
    }
}

extern "C" void kernel_launch(void* const* d_in, const int* in_sizes, int n_in,
                              void* d_out, int out_size, void* d_ws, size_t ws_size,
                              hipStream_t stream) {
    (void)n_in; (void)d_ws; (void)ws_size;

    const float* x          = (const float*)d_in[0];
    const int*   edge_index = (const int*)d_in[1];   // harness delivers ints as int32
    float*       out        = (float*)d_out;

    const int num_edges = in_sizes[1] / 2;           // edge_index is [2, E] flat

    // 1) Zero the (poisoned) output buffer, 16B at a time.
    {
        int n4 = out_size / 4;                       // N_NODES*32 is divisible by 4
        int threads = 256;
        int blocks = (n4 + threads - 1) / threads;
        zero_out_f4<<<blocks, threads, 0, stream>>>((float4*)out, n4);
    }

    // 2) Gather + atomic scatter-add. 8 threads per edge-pair lane-group,
    //    2 edges per group -> 4 threads per edge overall.
    {
        long long num_pairs     = ((long long)num_edges + 1) / 2;
        long long total_threads = num_pairs * 8;
        int threads = 256;
        long long blocks = (total_threads + threads - 1) / threads;
        scatter_add_edges<<<(unsigned int)blocks, threads, 0, stream>>>(
            x, edge_index, out, num_edges);
    }
}